// GNN_30459908063653
// MI455X (gfx1250) — compile-verified
//
#include <hip/hip_runtime.h>
#include <math.h>

typedef __attribute__((ext_vector_type(2))) float v2f;
typedef __attribute__((ext_vector_type(8))) float v8f;

#define F_IN 128
#define F_HID 32
#define F_OUT 2

// ---------------------------------------------------------------------------
// 1) init: deg = 1.0 (self loop), agg1 = 0, agg2 = 0
// ---------------------------------------------------------------------------
__global__ void init_kernel(float* __restrict__ deg, float* __restrict__ agg1,
                            float* __restrict__ agg2, int N) {
  long long tid = (long long)blockIdx.x * blockDim.x + threadIdx.x;
  long long total = (long long)N * F_HID;
  if (tid < total) agg1[tid] = 0.0f;
  if (tid < (long long)N * F_OUT) agg2[tid] = 0.0f;
  if (tid < N) deg[tid] = 1.0f;  // self-loop contribution to in-degree
}

// ---------------------------------------------------------------------------
// 2) in-degree with f32 atomics (stays in L2; deg buffer is 400 KB)
// ---------------------------------------------------------------------------
__global__ void degree_kernel(const int* __restrict__ col, float* __restrict__ deg,
                              int E) {
  int e = blockIdx.x * blockDim.x + threadIdx.x;
  if (e < E) atomicAdd(&deg[col[e]], 1.0f);
}

// ---------------------------------------------------------------------------
// 3) dinv = rsqrt(deg)   (deg >= 1 always, due to self loops)
// ---------------------------------------------------------------------------
__global__ void rsqrt_kernel(float* __restrict__ deg, int N) {
  int i = blockIdx.x * blockDim.x + threadIdx.x;
  if (i < N) deg[i] = rsqrtf(deg[i]);
}

// ---------------------------------------------------------------------------
// 4) h1 = x @ W1  via V_WMMA_F32_16X16X4_F32 (full f32 precision)
//    One wave produces a 16x32 output tile (two 16x16 WMMA accumulators).
//    4 waves / block; W1 (128x32 = 16 KB) staged in LDS.
//
//    ISA 32-bit A 16x4 layout: lanes 0-15 hold M=lane, VGPR0=K0,VGPR1=K1;
//    lanes 16-31 hold M=lane-16, VGPR0=K2,VGPR1=K3. B (4x16) mirrored.
//    C/D layout: VGPR i -> M=i (lanes 0-15, N=lane) / M=8+i (lanes 16-31).
// ---------------------------------------------------------------------------
__global__ __launch_bounds__(128) void gemm1_wmma_kernel(
    const float* __restrict__ x, const float* __restrict__ W1,
    float* __restrict__ h1, int N, int mtiles) {
  __shared__ float sW1[F_IN * F_HID];
  for (int i = threadIdx.x; i < F_IN * F_HID; i += 128) sW1[i] = W1[i];
  __syncthreads();

  const int wave = threadIdx.x >> 5;
  const int lane = threadIdx.x & 31;
  const int tile = blockIdx.x * 4 + wave;
  if (tile >= mtiles) return;  // uniform per wave: EXEC all-1 inside WMMA

  const int m0   = tile * 16;
  const int half = lane >> 4;          // 0: lanes 0-15, 1: lanes 16-31
  const int koff = half << 1;          // K sub-offset 0 or 2
  const int mrow = lane & 15;
  int rowA = m0 + mrow;
  if (rowA >= N) rowA = N - 1;         // safety clamp (N % 16 == 0 here)

  const float* __restrict__ xrow = x + (long long)rowA * F_IN + koff;

  v8f c0 = {};
  v8f c1 = {};
  for (int k0 = 0; k0 < F_IN; k0 += 4) {
    // A fragment: two consecutive K values for this lane's row
    float2 a2 = *(const float2*)(xrow + k0);
    v2f a; a.x = a2.x; a.y = a2.y;
    // B fragments from LDS, tile n0=0 and n0=16
    const int kb = k0 + koff;
    v2f b0, b1v;
    b0.x  = sW1[kb * F_HID + mrow];
    b0.y  = sW1[(kb + 1) * F_HID + mrow];
    b1v.x = sW1[kb * F_HID + 16 + mrow];
    b1v.y = sW1[(kb + 1) * F_HID + 16 + mrow];
    c0 = __builtin_amdgcn_wmma_f32_16x16x4_f32(false, a, false, b0,
                                               (short)0, c0, false, false);
    c1 = __builtin_amdgcn_wmma_f32_16x16x4_f32(false, a, false, b1v,
                                               (short)0, c1, false, false);
  }

  const int mBase = m0 + (half << 3);  // lanes >= 16 write rows M=8..15
#pragma unroll
  for (int i = 0; i < 8; ++i) {
    const int m = mBase + i;
    if (m < N) {
      h1[(long long)m * F_HID + mrow]      = c0[i];
      h1[(long long)m * F_HID + 16 + mrow] = c1[i];
    }
  }
}

// ---------------------------------------------------------------------------
// 5) edge scatter for F=32: one wave per edge, lane = feature.
//    Gather h1[row] (coalesced 128B), scale by norm, atomic-add into agg1[col].
// ---------------------------------------------------------------------------
__global__ __launch_bounds__(256) void edge_agg32_kernel(
    const int* __restrict__ row, const int* __restrict__ col,
    const float* __restrict__ dinv, const float* __restrict__ h1,
    float* __restrict__ agg1, int E) {
  long long t = (long long)blockIdx.x * blockDim.x + threadIdx.x;
  int e = (int)(t >> 5);
  int f = (int)(t & 31);
  if (e < E) {
    int r = row[e];
    int c = col[e];
    float nrm = dinv[r] * dinv[c];  // same-address loads broadcast in HW
    atomicAdd(&agg1[(long long)c * F_HID + f], nrm * h1[(long long)r * F_HID + f]);
  }
}

// ---------------------------------------------------------------------------
// 6) hrelu = relu(agg1 + dinv^2 * h1 + b1)   (in place over agg1)
// ---------------------------------------------------------------------------
__global__ void post1_kernel(float* __restrict__ agg1, const float* __restrict__ h1,
                             const float* __restrict__ dinv,
                             const float* __restrict__ b1, int N) {
  long long tid = (long long)blockIdx.x * blockDim.x + threadIdx.x;
  if (tid < (long long)N * F_HID) {
    int i = (int)(tid >> 5);
    int f = (int)(tid & 31);
    float di = dinv[i];
    float v = agg1[tid] + di * di * h1[tid] + b1[f];
    agg1[tid] = fmaxf(v, 0.0f);
  }
}

// ---------------------------------------------------------------------------
// 7) h2 = hrelu @ W2  (32 -> 2): scalar, one thread per node; W2 is uniform
//    so its loads become scalar loads and stay in K$/L2.
// ---------------------------------------------------------------------------
__global__ void gemm2_kernel(const float* __restrict__ hrelu,
                             const float* __restrict__ W2,
                             float* __restrict__ h2, int N) {
  int i = blockIdx.x * blockDim.x + threadIdx.x;
  if (i < N) {
    const float4* hr = (const float4*)(hrelu + (long long)i * F_HID);
    float acc0 = 0.0f, acc1 = 0.0f;
#pragma unroll
    for (int q = 0; q < F_HID / 4; ++q) {
      float4 v = hr[q];
      int f = q * 4;
      acc0 += v.x * W2[(f + 0) * 2 + 0] + v.y * W2[(f + 1) * 2 + 0] +
              v.z * W2[(f + 2) * 2 + 0] + v.w * W2[(f + 3) * 2 + 0];
      acc1 += v.x * W2[(f + 0) * 2 + 1] + v.y * W2[(f + 1) * 2 + 1] +
              v.z * W2[(f + 2) * 2 + 1] + v.w * W2[(f + 3) * 2 + 1];
    }
    h2[(long long)i * 2 + 0] = acc0;
    h2[(long long)i * 2 + 1] = acc1;
  }
}

// ---------------------------------------------------------------------------
// 8) edge scatter for F=2: one thread per edge (8B gather, 2 atomics)
// ---------------------------------------------------------------------------
__global__ void edge_agg2_kernel(const int* __restrict__ row,
                                 const int* __restrict__ col,
                                 const float* __restrict__ dinv,
                                 const float* __restrict__ h2,
                                 float* __restrict__ agg2, int E) {
  int e = blockIdx.x * blockDim.x + threadIdx.x;
  if (e < E) {
    int r = row[e];
    int c = col[e];
    float nrm = dinv[r] * dinv[c];
    float2 v = *(const float2*)(h2 + (long long)r * 2);
    atomicAdd(&agg2[(long long)c * 2 + 0], nrm * v.x);
    atomicAdd(&agg2[(long long)c * 2 + 1], nrm * v.y);
  }
}

// ---------------------------------------------------------------------------
// 9) out = log_softmax(agg2 + dinv^2 * h2 + b2)   over the 2 classes
// ---------------------------------------------------------------------------
__global__ void final_kernel(const float* __restrict__ agg2,
                             const float* __restrict__ h2,
                             const float* __restrict__ dinv,
                             const float* __restrict__ b2,
                             float* __restrict__ out, int N) {
  int i = blockIdx.x * blockDim.x + threadIdx.x;
  if (i < N) {
    float di = dinv[i];
    float d2 = di * di;
    float v0 = agg2[(long long)i * 2 + 0] + d2 * h2[(long long)i * 2 + 0] + b2[0];
    float v1 = agg2[(long long)i * 2 + 1] + d2 * h2[(long long)i * 2 + 1] + b2[1];
    float m = fmaxf(v0, v1);
    float lse = m + logf(expf(v0 - m) + expf(v1 - m));
    out[(long long)i * 2 + 0] = v0 - lse;
    out[(long long)i * 2 + 1] = v1 - lse;
  }
}

// ---------------------------------------------------------------------------
// launch
// ---------------------------------------------------------------------------
extern "C" void kernel_launch(void* const* d_in, const int* in_sizes, int n_in,
                              void* d_out, int out_size, void* d_ws, size_t ws_size,
                              hipStream_t stream) {
  const float* x  = (const float*)d_in[0];
  const int*   ei = (const int*)d_in[1];
  const float* W1 = (const float*)d_in[2];
  const float* b1 = (const float*)d_in[3];
  const float* W2 = (const float*)d_in[4];
  const float* b2 = (const float*)d_in[5];
  float* out = (float*)d_out;

  const int N = in_sizes[0] / F_IN;
  const int E = in_sizes[1] / 2;
  const int* row = ei;       // edge_index[0]
  const int* col = ei + E;   // edge_index[1]

  // workspace layout (floats): dinv[N] | h1[32N] | agg1/hrelu[32N] | h2[2N] | agg2[2N]
  float* ws   = (float*)d_ws;
  float* dinv = ws;
  float* h1   = dinv + N;
  float* agg1 = h1 + (size_t)32 * N;
  float* h2   = agg1 + (size_t)32 * N;
  float* agg2 = h2 + (size_t)2 * N;

  const long long nh = (long long)N * F_HID;

  init_kernel<<<(int)((nh + 255) / 256), 256, 0, stream>>>(dinv, agg1, agg2, N);
  degree_kernel<<<(E + 255) / 256, 256, 0, stream>>>(col, dinv, E);
  rsqrt_kernel<<<(N + 255) / 256, 256, 0, stream>>>(dinv, N);

  const int mtiles = (N + 15) / 16;
  gemm1_wmma_kernel<<<(mtiles + 3) / 4, 128, 0, stream>>>(x, W1, h1, N, mtiles);

  const long long e32 = (long long)E * 32;
  edge_agg32_kernel<<<(int)((e32 + 255) / 256), 256, 0, stream>>>(row, col, dinv,
                                                                  h1, agg1, E);
  post1_kernel<<<(int)((nh + 255) / 256), 256, 0, stream>>>(agg1, h1, dinv, b1, N);
  gemm2_kernel<<<(N + 255) / 256, 256, 0, stream>>>(agg1, W2, h2, N);
  edge_agg2_kernel<<<(E + 255) / 256, 256, 0, stream>>>(row, col, dinv, h2, agg2, E);
  final_kernel<<<(N + 255) / 256, 256, 0, stream>>>(agg2, h2, dinv, b2, out, N);
}